// SparseProp_47665547051029
// MI455X (gfx1250) — compile-verified
//
#include <hip/hip_runtime.h>
#include <stdint.h>

// GCN propagation: out = D^-1/2 (A + A^T) D^-1/2 X
//   N = 100000 nodes, D_FEAT = 128, E = 300000 undirected edges (600000 directed)
// Memory-bound gather/scatter: x (51MB) + out (51MB) fit in MI455X's 192MB L2,
// so edge gathers and float atomics resolve in L2. CDNA5 paths used:
//  - per-lane async global->LDS B128 copies (ASYNCcnt, double buffered per wave)
//  - s_wait_asynccnt split-counter waits
//  - ds_load_b128 readback (explicit addrspace(3)), global f32 atomics scatter
//  - scalar (readfirstlane-uniformized) edge-index/norm fetches via SMEM
// One directed edge == one wave-iteration: 128 f32 = 32 lanes x 16B (wave32).

#define D_FEAT 128
#define BLOCK 256
#define WAVES_PER_BLOCK (BLOCK / 32)

typedef int   v4i __attribute__((ext_vector_type(4)));
typedef float v4f __attribute__((ext_vector_type(4)));
typedef __attribute__((address_space(1))) v4i gv4i;   // global v4i32
typedef __attribute__((address_space(3))) v4i lv4i;   // LDS v4i32
typedef __attribute__((address_space(3))) v4f lv4f;   // LDS v4f32

#if defined(__AMDGCN__) && __has_builtin(__builtin_amdgcn_global_load_async_to_lds_b128)
#define HAVE_ASYNC_LDS 1
#else
#define HAVE_ASYNC_LDS 0
#if defined(__AMDGCN__)
#warning "gfx1250 __builtin_amdgcn_global_load_async_to_lds_b128 not found; using direct-load fallback"
#endif
#endif

#if defined(__AMDGCN__) && __has_builtin(__builtin_amdgcn_s_wait_asynccnt)
#define WAIT_ASYNC(n) __builtin_amdgcn_s_wait_asynccnt(n)
#elif defined(__AMDGCN__)
#define WAIT_ASYNC(n) asm volatile("s_wait_asynccnt %0" ::"n"(n) : "memory")
#else
#define WAIT_ASYNC(n)
#endif

#if defined(__AMDGCN__)
#define UNIFORM(x) __builtin_amdgcn_readfirstlane(x)
#else
#define UNIFORM(x) (x)
#endif

// ------------------------------------------------------------------- SpMM
// One directed edge per wave-iteration. Async-gather x[col] (512B) into a
// per-wave LDS double buffer, then scale + scatter with f32 atomics.
__global__ __launch_bounds__(BLOCK) void spmm_kernel(const int* __restrict__ esrc,
                                                     const int* __restrict__ edst,
                                                     const float* __restrict__ rs,
                                                     const float* __restrict__ x,
                                                     float* __restrict__ out, int E) {
    __shared__ __align__(16) float tile[WAVES_PER_BLOCK * 2 * D_FEAT];  // 8 KB

    const int lane    = threadIdx.x & 31;
    const int wave    = threadIdx.x >> 5;
    const int twoE    = 2 * E;
    const int wstride = gridDim.x * WAVES_PER_BLOCK;
    // wave-uniform edge id, made *provably* uniform so index fetches go SMEM
    int e = UNIFORM(blockIdx.x * WAVES_PER_BLOCK + wave);
    if (e >= twoE) return;

    float* buf[2] = { &tile[(wave * 2 + 0) * D_FEAT],
                      &tile[(wave * 2 + 1) * D_FEAT] };

    auto load_edge = [&](int ed, int& r, int& c) {
        // prefetch the index stream a few iterations ahead (global_prefetch_b8)
        __builtin_prefetch(esrc + (ed < E ? ed : ed - E) + 4 * wstride, 0, 3);
        if (ed < E) { r = esrc[ed];     c = edst[ed];     }
        else        { r = edst[ed - E]; c = esrc[ed - E]; }
    };

#if HAVE_ASYNC_LDS
    auto issue = [&](int c, float* b) {
        const float* g = x + (size_t)c * D_FEAT + lane * 4;
        __builtin_amdgcn_global_load_async_to_lds_b128(
            (gv4i*)g, (lv4i*)(b + lane * 4), /*offset=*/0, /*cpol=*/0);
    };

    int row, col;
    load_edge(e, row, col);
    issue(col, buf[0]);
    int cur = 0;
    for (;;) {
        const int en = UNIFORM(e + wstride);
        const bool more = en < twoE;          // scalar branch
        int rown = 0, coln = 0;
        if (more) {
            load_edge(en, rown, coln);
            issue(coln, buf[cur ^ 1]);
            WAIT_ASYNC(1);                     // in-order: current buffer landed
        } else {
            WAIT_ASYNC(0);
        }
        const float val = rs[row] * rs[col];
        // explicit addrspace(3) read -> ds_load_b128 (not flat)
        const v4f v = *(lv4f*)(buf[cur] + lane * 4);
        float* o = out + (size_t)row * D_FEAT + lane * 4;
        atomicAdd(o + 0, v.x * val);
        atomicAdd(o + 1, v.y * val);
        atomicAdd(o + 2, v.z * val);
        atomicAdd(o + 3, v.w * val);
        if (!more) break;
        e = en; row = rown; col = coln; cur ^= 1;
    }
#else
    for (; e < twoE; e = UNIFORM(e + wstride)) {
        int row, col;
        load_edge(e, row, col);
        const float val = rs[row] * rs[col];
        const float4 v =
            *reinterpret_cast<const float4*>(x + (size_t)col * D_FEAT + lane * 4);
        float* o = out + (size_t)row * D_FEAT + lane * 4;
        atomicAdd(o + 0, v.x * val);
        atomicAdd(o + 1, v.y * val);
        atomicAdd(o + 2, v.z * val);
        atomicAdd(o + 3, v.w * val);
    }
#endif
}

// ---------------------------------------------------------------- zero fill
__global__ __launch_bounds__(BLOCK) void zero_f32x4(float* __restrict__ p, int n4) {
    int i = blockIdx.x * BLOCK + threadIdx.x;
    if (i < n4) reinterpret_cast<float4*>(p)[i] = make_float4(0.f, 0.f, 0.f, 0.f);
}

// ------------------------------------------------------------------ degrees
// deg[i] = #incident directed-edge endpoints (both orientations), via L2 atomics
__global__ __launch_bounds__(BLOCK) void degree_kernel(const int* __restrict__ esrc,
                                                       const int* __restrict__ edst,
                                                       float* __restrict__ deg, int E) {
    int e = blockIdx.x * BLOCK + threadIdx.x;
    if (e < E) {
        atomicAdd(&deg[esrc[e]], 1.0f);
        atomicAdd(&deg[edst[e]], 1.0f);
    }
}

// deg[i] <- rsqrt(max(deg[i],1))
__global__ __launch_bounds__(BLOCK) void rsqrt_kernel(float* __restrict__ deg, int N) {
    int i = blockIdx.x * BLOCK + threadIdx.x;
    if (i < N) deg[i] = rsqrtf(fmaxf(deg[i], 1.0f));
}

// ---------------------------------------------------------------- launcher
extern "C" void kernel_launch(void* const* d_in, const int* in_sizes, int n_in,
                              void* d_out, int out_size, void* d_ws, size_t ws_size,
                              hipStream_t stream) {
    const float* x     = (const float*)d_in[0];     // [N, 128] f32
    const int*   edges = (const int*)d_in[1];       // [2, E] int32 row-major
    const int    E     = in_sizes[1] / 2;
    const int    N     = in_sizes[0] / D_FEAT;
    const int*   esrc  = edges;
    const int*   edst  = edges + E;
    float* out = (float*)d_out;
    float* deg = (float*)d_ws;                      // N floats of scratch

    // zero accumulators every call (atomics accumulate; replays don't re-poison)
    {
        int n4 = out_size / 4;
        zero_f32x4<<<(n4 + BLOCK - 1) / BLOCK, BLOCK, 0, stream>>>(out, n4);
        int d4 = (N + 3) / 4;
        zero_f32x4<<<(d4 + BLOCK - 1) / BLOCK, BLOCK, 0, stream>>>(deg, d4);
    }
    degree_kernel<<<(E + BLOCK - 1) / BLOCK, BLOCK, 0, stream>>>(esrc, edst, deg, E);
    rsqrt_kernel<<<(N + BLOCK - 1) / BLOCK, BLOCK, 0, stream>>>(deg, N);

    // 4096 blocks x 8 waves = 32768 waves; ~18 directed edges per wave,
    // enough loop trips to keep the async double buffer pipelined.
    spmm_kernel<<<4096, BLOCK, 0, stream>>>(esrc, edst, deg, x, out, E);
}